// SDM_48077863912068
// MI455X (gfx1250) — compile-verified
//
#include <hip/hip_runtime.h>

typedef __attribute__((ext_vector_type(2))) float v2f;
typedef __attribute__((ext_vector_type(8))) float v8f;

#define NUM_NODE   256
#define INPUT_DIM  256
#define LATENT     8192
#define OUT_DIM    2048
#define TOPK_ITERS 256
#define KSTAGE     32   // K-rows staged per LDS buffer

// ---------------------------------------------------------------------------
// Cooperative  Out = relu(A[M,K] @ W[K,N] + bias[N])  with V_WMMA_F32_16X16X4_F32.
//
// Block = 256 threads (8 waves). Block tile = 128(M) x 64(N); wave w owns a
// 16-row M-subtile. The shared B tile (32 K-rows x 64 N) is staged into LDS
// with global_load_async_to_lds_b32 (ASYNCcnt), double buffered.
//
// LDS layout is K-pair swizzled:  Bs[p*128 + n*2 + (k&1)],  p = k>>1, so each
// B fragment {B[k][n], B[k+1][n]} is one contiguous 8B LDS read (ds_load_b64 /
// ds_load_2addr_b64, no re-pack movs).
//
// Fragment layouts (wave32, CDNA5 ISA 7.12.2):
//   A 16x4 : lane0-15 -> M=lane, K={0,1}; lane16-31 -> K={2,3}
//   B 4x16 : lane0-15 -> N=lane, K={0,1}; lane16-31 -> K={2,3}
//   C/D    : 8 VGPRs; N = lane&15, M = vgpr + (lane>>4)*8
// ---------------------------------------------------------------------------
__global__ __launch_bounds__(256) void gemm_relu_wmma_f32(
    const float* __restrict__ A, const float* __restrict__ W,
    const float* __restrict__ bias, float* __restrict__ Out,
    int M, int K, int N)
{
    __shared__ float Bs[2][KSTAGE * 64];

    const int tid  = threadIdx.x;
    const int wv   = tid >> 5;             // wave in block: 0..7
    const int lane = tid & 31;

    const int nblk = N >> 6;               // 64-wide N tiles
    const int bm   = blockIdx.x / nblk;    // 128-row M block
    const int bn   = blockIdx.x % nblk;
    const int n0   = bn << 6;
    const int m0   = (bm << 7) + (wv << 4);

    const int l15 = lane & 15;             // M index (A) / N index (B,C,D)
    const int kg  = (lane >> 4) << 1;      // K pair base: 0 or 2

    // ---- staging map: element e = j*256 + tid (j=0..7) of the 32x64 tile ----
    //   stage row r = j*4 + rhi, col n = nlo  (lanes of one wave: 32 consecutive n)
    //   LDS bytes:  p*512 + n*8 + q*4,  p = r>>1, q = r&1
    const int rhi = tid >> 6;              // 0..3
    const int nlo = tid & 63;
    const int phi = rhi >> 1;              // contribution to p
    const int qb  = rhi & 1;               // parity bit

    #define STAGE_ISSUE(sk, bufbase)                                                    \
        do {                                                                            \
            unsigned lbase = (unsigned)(uintptr_t)(bufbase);                            \
            _Pragma("unroll")                                                           \
            for (int j = 0; j < 8; ++j) {                                               \
                unsigned goff = (unsigned)(((((sk) + (j << 2) + rhi) * N) + n0 + nlo) * 4); \
                unsigned l0   = lbase + (unsigned)(((j << 1) + phi) * 512 + nlo * 8 + qb * 4); \
                asm volatile("global_load_async_to_lds_b32 %0, %1, %2 offset:0"         \
                             :: "v"(l0), "v"(goff), "s"(W) : "memory");                 \
            }                                                                           \
        } while (0)

    v8f acc0 = {}, acc1 = {}, acc2 = {}, acc3 = {};
    const float* arow = A + (size_t)(m0 + l15) * K;

    const int nstages = K / KSTAGE;
    STAGE_ISSUE(0, &Bs[0][0]);             // prefetch stage 0

    for (int s = 0; s < nstages; ++s) {
        if (s + 1 < nstages) {
            if ((s + 1) & 1) STAGE_ISSUE((s + 1) * KSTAGE, &Bs[1][0]);
            else             STAGE_ISSUE((s + 1) * KSTAGE, &Bs[0][0]);
            asm volatile("s_wait_asynccnt 0x8" ::: "memory");  // stage s complete
        } else {
            asm volatile("s_wait_asynccnt 0x0" ::: "memory");
        }
        __syncthreads();                   // stage s visible to all waves

        const float* bsb = &Bs[s & 1][0];
        const int    kb  = s * KSTAGE;
        // fragment base in swizzled tile: p = kk/2 + kg/2, float idx p*128 + l15*2
        const float* bp0 = bsb + (kg >> 1) * 128 + l15 * 2;

        // software pipeline: fetch step kk+4 before WMMAs of step kk
        v2f a  = *(const v2f*)(arow + kb + kg);
        v2f b0 = *(const v2f*)(bp0 + 0);
        v2f b1 = *(const v2f*)(bp0 + 32);
        v2f b2 = *(const v2f*)(bp0 + 64);
        v2f b3 = *(const v2f*)(bp0 + 96);

#pragma unroll
        for (int kk = 0; kk < KSTAGE; kk += 4) {
            v2f an = a, c0 = b0, c1 = b1, c2 = b2, c3 = b3;
            if (kk + 4 < KSTAGE) {
                an = *(const v2f*)(arow + kb + kk + 4 + kg);
                const float* np = bp0 + ((kk + 4) >> 1) * 128;
                c0 = *(const v2f*)(np + 0);
                c1 = *(const v2f*)(np + 32);
                c2 = *(const v2f*)(np + 64);
                c3 = *(const v2f*)(np + 96);
            }
            acc0 = __builtin_amdgcn_wmma_f32_16x16x4_f32(false, a, false, b0, (short)0, acc0, false, false);
            acc1 = __builtin_amdgcn_wmma_f32_16x16x4_f32(false, a, false, b1, (short)0, acc1, false, false);
            acc2 = __builtin_amdgcn_wmma_f32_16x16x4_f32(false, a, false, b2, (short)0, acc2, false, false);
            acc3 = __builtin_amdgcn_wmma_f32_16x16x4_f32(false, a, false, b3, (short)0, acc3, false, false);
            a = an; b0 = c0; b1 = c1; b2 = c2; b3 = c3;
        }
        __syncthreads();                   // all waves done with this buffer
    }
    #undef STAGE_ISSUE

    // Bias + ReLU + store. D layout: N = lane&15, M = v + (lane>>4)*8.
    const int   mb  = (lane >> 4) << 3;
    const float bb0 = bias[n0 + l15];
    const float bb1 = bias[n0 + 16 + l15];
    const float bb2 = bias[n0 + 32 + l15];
    const float bb3 = bias[n0 + 48 + l15];
#pragma unroll
    for (int v = 0; v < 8; ++v) {
        float* orow = Out + (size_t)(m0 + mb + v) * N + n0 + l15;
        orow[0]  = fmaxf(acc0[v] + bb0, 0.0f);
        orow[16] = fmaxf(acc1[v] + bb1, 0.0f);
        orow[32] = fmaxf(acc2[v] + bb2, 0.0f);
        orow[48] = fmaxf(acc3[v] + bb3, 0.0f);
    }
}

// ---------------------------------------------------------------------------
// Top_M: relaxed top-k via iterative Gumbel-softmax. One 256-thread block per
// row; keys/onehot/khot in registers (8 cols/thread, stride 256). Row max/sum
// via wave32 shfl-xor butterfly + 8-slot LDS cross-wave reduce (2 barriers/it).
// ---------------------------------------------------------------------------
__global__ __launch_bounds__(256) void topm_kernel(
    const float* __restrict__ ZA, const float* __restrict__ Z,
    float* __restrict__ Out)
{
    const int row  = blockIdx.x;
    const int tid  = threadIdx.x;
    const int wid  = tid >> 5;
    const int lane = tid & 31;

    const float* za = ZA + (size_t)row * OUT_DIM;
    const float* zg = Z  + (size_t)row * OUT_DIM;

    float keys[8], onehot[8], khot[8];
#pragma unroll
    for (int j = 0; j < 8; ++j) {
        const int c = tid + (j << 8);
        keys[j]   = __logf(za[c] + 1e-30f) + zg[c];
        onehot[j] = 0.0f;
        khot[j]   = 0.0f;
    }

    __shared__ float redmax[8];
    __shared__ float redsum[8];

    for (int it = 0; it < TOPK_ITERS; ++it) {
        // 1) keys += log(max(1 - onehot, MASK_EPS)); local max
        float m = -3.4e38f;
#pragma unroll
        for (int j = 0; j < 8; ++j) {
            keys[j] += __logf(fmaxf(1.0f - onehot[j], 1e-20f));
            m = fmaxf(m, keys[j]);
        }
        // 2) row max
#pragma unroll
        for (int off = 16; off > 0; off >>= 1)
            m = fmaxf(m, __shfl_xor(m, off, 32));
        if (lane == 0) redmax[wid] = m;
        __syncthreads();
        m = redmax[0];
#pragma unroll
        for (int w = 1; w < 8; ++w) m = fmaxf(m, redmax[w]);

        // 3) exp((keys - max)/T), row sum  (TEMP = 2.0)
        float s = 0.0f;
#pragma unroll
        for (int j = 0; j < 8; ++j) {
            float e = __expf((keys[j] - m) * 0.5f);
            onehot[j] = e;
            s += e;
        }
#pragma unroll
        for (int off = 16; off > 0; off >>= 1)
            s += __shfl_xor(s, off, 32);
        if (lane == 0) redsum[wid] = s;
        __syncthreads();
        s = redsum[0];
#pragma unroll
        for (int w = 1; w < 8; ++w) s += redsum[w];

        // 4) normalize, accumulate khot
        const float inv = __frcp_rn(s);
#pragma unroll
        for (int j = 0; j < 8; ++j) {
            onehot[j] *= inv;
            khot[j]   += onehot[j];
        }
    }

    float* out = Out + (size_t)row * OUT_DIM;
#pragma unroll
    for (int j = 0; j < 8; ++j)
        out[tid + (j << 8)] = khot[j];
}

extern "C" void kernel_launch(void* const* d_in, const int* in_sizes, int n_in,
                              void* d_out, int out_size, void* d_ws, size_t ws_size,
                              hipStream_t stream) {
    const float* adj = (const float*)d_in[0];   // [256, 256]
    const float* W1  = (const float*)d_in[1];   // [256, 8192]
    const float* b1  = (const float*)d_in[2];   // [8192]
    const float* W2  = (const float*)d_in[3];   // [8192, 2048]
    const float* b2  = (const float*)d_in[4];   // [2048]
    const float* z   = (const float*)d_in[5];   // [256, 2048]
    float* out = (float*)d_out;                 // [256, 2048]

    float* h  = (float*)d_ws;                   // [256, 8192]  (8 MB)
    float* za = h + (size_t)NUM_NODE * LATENT;  // [256, 2048]  (2 MB)

    // GEMM1: (256x256)x(256x8192): blocks = (256/128)*(8192/64) = 256
    gemm_relu_wmma_f32<<<(NUM_NODE / 128) * (LATENT / 64), 256, 0, stream>>>(
        adj, W1, b1, h, NUM_NODE, INPUT_DIM, LATENT);

    // GEMM2: (256x8192)x(8192x2048): blocks = (256/128)*(2048/64) = 64
    gemm_relu_wmma_f32<<<(NUM_NODE / 128) * (OUT_DIM / 64), 256, 0, stream>>>(
        h, W2, b2, za, NUM_NODE, LATENT, OUT_DIM);

    // Top_M: one block per row
    topm_kernel<<<NUM_NODE, 256, 0, stream>>>(za, z, out);
}